// BilateralFilter3D_49658411877033
// MI455X (gfx1250) — compile-verified
//
#include <hip/hip_runtime.h>
#include <hip/hip_bf16.h>

// Bilateral 3D filter, 3x3x3 window, replicate padding.
// Volume: (B=2, C=1, D=128, H=192, W=192) fp32.
// Strategy: async global->LDS halo tile (ASYNCcnt path); per-thread x-pair of
// voxels so tap math vectorizes to V_PK_*_F32; one v_exp_f32 per non-center
// tap (center tap weight == 1 exactly, so it is free).

#define W_   192
#define H_   192
#define D_   128
#define HW_  (H_ * W_)
#define DHW_ (D_ * HW_)

#define TX 64
#define TY 4
#define TZ 4
#define LX (TX + 2)          // 66
#define LY (TY + 2)          // 6
#define LZ (TZ + 2)          // 6
#define LPLANE (LY * LX)     // 396
#define LTOT   (LZ * LPLANE) // 2376

typedef __attribute__((address_space(1))) int as1_int;   // global
typedef __attribute__((address_space(3))) int as3_int;   // LDS
typedef float v2f __attribute__((ext_vector_type(2)));

__device__ __forceinline__ float fast_exp2(float x) {
#if __has_builtin(__builtin_amdgcn_exp2f)
    return __builtin_amdgcn_exp2f(x);
#else
    return exp2f(x);
#endif
}

__device__ __forceinline__ float fast_rcp(float x) {
#if __has_builtin(__builtin_amdgcn_rcpf)
    return __builtin_amdgcn_rcpf(x);
#else
    return 1.0f / x;
#endif
}

__device__ __forceinline__ v2f splat2(float x) { v2f r; r.x = x; r.y = x; return r; }

__global__ __launch_bounds__(256)
void BilateralFilter3D_49658411877033_kernel(const float* __restrict__ vol,
                                             float* __restrict__ out)
{
    __shared__ float tile[LTOT];

    const int tid = threadIdx.x;
    const int x0  = blockIdx.x * TX;
    const int y0  = blockIdx.y * TY;
    const int bz  = blockIdx.z;              // 0..63
    const int b   = bz >> 5;                 // batch (D_/TZ == 32 tiles per batch)
    const int z0  = (bz & 31) * TZ;
    const int base = b * DHW_;

    // ---- Stage halo tile into LDS via async loads (ASYNCcnt path) ----
    for (int i = tid; i < LTOT; i += 256) {
        const int lz = i / LPLANE;
        const int r  = i - lz * LPLANE;
        const int ly = r / LX;
        const int lx = r - ly * LX;
        const int gz = min(max(z0 - 1 + lz, 0), D_ - 1);
        const int gy = min(max(y0 - 1 + ly, 0), H_ - 1);
        const int gx = min(max(x0 - 1 + lx, 0), W_ - 1);
        const float* gp = vol + base + gz * HW_ + gy * W_ + gx;
#if __has_builtin(__builtin_amdgcn_global_load_async_to_lds_b32)
        __builtin_amdgcn_global_load_async_to_lds_b32(
            (as1_int*)gp, (as3_int*)(&tile[i]), 0, 0);
#else
        tile[i] = *gp;
#endif
    }
#if __has_builtin(__builtin_amdgcn_s_wait_asynccnt)
    __builtin_amdgcn_s_wait_asynccnt(0);
#endif
    __syncthreads();

    // ---- Compute: w = exp2(KR*d^2 + LSC*dist2), packed over an x-pair ----
    constexpr float KR  = -0.500935813f;     // -1/(2*1.2^2*ln2)
    constexpr float LSC = -5.00935813e-5f;   // -1/(2*120^2*ln2)

    const int px  = tid & 31;        // x pair index: voxels 2*px, 2*px+1
    const int ty  = (tid >> 5) & 3;  // y within tile
    const int tzh = tid >> 7;        // z half: layers {2*tzh, 2*tzh+1}
    const int lxc = 2 * px + 1;      // LDS x of first voxel's center

    const v2f KRv = splat2(KR);

#pragma unroll
    for (int zi = 0; zi < 2; ++zi) {
        const int z = 2 * tzh + zi;
        const int crow = (z + 1) * LPLANE + (ty + 1) * LX + lxc;
        v2f c; c.x = tile[crow]; c.y = tile[crow + 1];

        // Center tap: d == 0, dist2 == 0  =>  w == 1 exactly.
        v2f num = c;
        v2f den = splat2(1.0f);

#pragma unroll
        for (int dz = -1; dz <= 1; ++dz) {
#pragma unroll
            for (int dy = -1; dy <= 1; ++dy) {
                const int rbase = (z + 1 + dz) * LPLANE + (ty + 1 + dy) * LX + lxc - 1;
                const float r0 = tile[rbase + 0];
                const float r1 = tile[rbase + 1];
                const float r2 = tile[rbase + 2];
                const float r3 = tile[rbase + 3];
#pragma unroll
                for (int dx = -1; dx <= 1; ++dx) {
                    if (dz == 0 && dy == 0 && dx == 0) continue;  // center handled above
                    const float lsw = LSC * (float)(dz * dz + dy * dy + dx * dx);
                    v2f n;
                    n.x = (dx == -1) ? r0 : ((dx == 0) ? r1 : r2);
                    n.y = (dx == -1) ? r1 : ((dx == 0) ? r2 : r3);
                    const v2f d  = n - c;                                   // v_pk_add (neg)
                    const v2f kd = d * KRv;                                 // v_pk_mul
                    const v2f t  = __builtin_elementwise_fma(kd, d, splat2(lsw)); // v_pk_fma
                    v2f w; w.x = fast_exp2(t.x); w.y = fast_exp2(t.y);      // 2x v_exp_f32
                    num = __builtin_elementwise_fma(w, n, num);             // v_pk_fma
                    den = den + w;                                          // v_pk_add
                }
            }
        }
        // den >= 1 (center weight), so the reference's clip(1e-8) is identity.
        v2f res;
        res.x = num.x * fast_rcp(den.x);
        res.y = num.y * fast_rcp(den.y);
        // 8-byte aligned packed store (x0 + 2*px is even)
        *(v2f*)(out + base + (z0 + z) * HW_ + (y0 + ty) * W_ + (x0 + 2 * px)) = res;
    }
}

extern "C" void kernel_launch(void* const* d_in, const int* in_sizes, int n_in,
                              void* d_out, int out_size, void* d_ws, size_t ws_size,
                              hipStream_t stream) {
    (void)in_sizes; (void)n_in; (void)d_ws; (void)ws_size; (void)out_size;
    const float* vol = (const float*)d_in[0];
    float* out = (float*)d_out;
    dim3 grid(W_ / TX, H_ / TY, (D_ / TZ) * 2);   // (3, 48, 64)
    BilateralFilter3D_49658411877033_kernel<<<grid, dim3(256), 0, stream>>>(vol, out);
}